// YOLOLoss_69166153335192
// MI455X (gfx1250) — compile-verified
//
#include <hip/hip_runtime.h>
#include <hip/hip_bf16.h>

// ---------------------------------------------------------------------------
// YOLO head loss for MI455X (gfx1250, wave32).
// Latency-bound gather+reduce problem (~1.6 MB gathered, ~10 MFLOP), so:
//   stage 1: one wave handles 16 (b,m) pairs; the 80-class focal reduction is
//            done with V_WMMA_F32_16X16X4_F32 (A = focal terms, B = ones,
//            C = running accumulator over 3 levels x 20 K-chunks).
//   stage 2: single-wave deterministic finalize over 13 block partials.
// Focal math: one v_exp_f32 + one raw v_log_f32 + one raw v_rcp_f32 per
// class; all selects are float cndmasks so EXEC stays all-1s through every
// WMMA.  Raw hardware trans ops are safe because their arguments are
// confined to (1,2] (no denormals, no overflow).
// ---------------------------------------------------------------------------

typedef __attribute__((ext_vector_type(2))) float v2f;
typedef __attribute__((ext_vector_type(8))) float v8f;

#define NUM_CLASSES 80
#define BB 32
#define MM 50
#define NN 8400
#define PAIRS (BB * MM)                       // 1600
#define WAVES_PER_BLOCK 8
#define BLOCK_THREADS (WAVES_PER_BLOCK * 32)  // 256 = 8 wave32s
#define PAIRS_PER_BLOCK (WAVES_PER_BLOCK * 16)
#define NUM_BLOCKS ((PAIRS + PAIRS_PER_BLOCK - 1) / PAIRS_PER_BLOCK)  // 13

#define LOG2E 1.4426950408889634f
#define LN2   0.6931471805599453f

// Focal-BCE term for one logit.
//   e = exp(-|x|) = exp2(-|x|*log2e)        in (0,1]
//   r = log1p(e)  = ln2 * log2(1+e)         arg in (1,2] -> raw v_log_f32 ok
//   softplus(x)  = relu(x)  + r   (ce, t=0)
//   softplus(-x) = relu(-x) + r   (ce, t=1)
//   p = (x>0 ? 1 : e) * rcp(1+e)            raw v_rcp_f32 ok (arg in (1,2])
__device__ __forceinline__ float focal_term(float x, bool is_t) {
    const float e      = __builtin_amdgcn_exp2f(-fabsf(x) * LOG2E);
    const float ope    = 1.0f + e;
    const float r      = LN2 * __builtin_amdgcn_logf(ope);
    const float sp_pos = fmaxf(-x, 0.0f) + r;   // -log sigmoid(x)
    const float sp_neg = fmaxf( x, 0.0f) + r;   // -log sigmoid(-x)
    const float inv1pe = __builtin_amdgcn_rcpf(ope);
    const float num    = (x > 0.0f) ? 1.0f : e;
    const float p      = num * inv1pe;          // sigmoid(x)
    const float ce     = is_t ? sp_pos : sp_neg;
    const float omp    = is_t ? (1.0f - p) : p; // 1 - p_t
    const float at     = is_t ? 0.25f : 0.75f;  // alpha_t
    return at * omp * omp * ce;                 // gamma = 2
}

__global__ __launch_bounds__(BLOCK_THREADS)
void yolo_loss_stage1(const float* __restrict__ box_preds,   // [B,N,4]
                      const float* __restrict__ cls_preds,   // [B,N,80]
                      const float* __restrict__ gt_boxes,    // [B,M,4]
                      const int*   __restrict__ gt_labels,   // [B,M]
                      const unsigned char* __restrict__ gt_mask, // [B,M] bool8
                      float* __restrict__ partials) {        // [grid][3]
    const int lane  = threadIdx.x & 31;
    const int wid   = threadIdx.x >> 5;
    const int gwave = blockIdx.x * WAVES_PER_BLOCK + wid;
    const int half  = lane >> 4;   // lanes 0-15 vs 16-31 share a pair
    const int sub   = lane & 15;

    int q = gwave * 16 + sub;                 // pair id = b*M + m
    const bool pair_ok = (q < PAIRS);
    q = pair_ok ? q : 0;                      // safe clamp for tail waves
    const int b = q / MM;

    const float4 gb    = *(const float4*)(gt_boxes + (size_t)q * 4);
    const int    label = gt_labels[q];
    const bool   gmask = pair_ok && (gt_mask[q] != 0);
    const float  cx    = (gb.x + gb.z) * 0.5f;
    const float  cy    = (gb.y + gb.w) * 0.5f;
    const float  garea = (gb.z - gb.x) * (gb.w - gb.y);
    const float  gate  = (half == 0) ? 1.0f : 0.0f;  // count box/pos once

    v8f cacc = {};                     // WMMA accumulator (cls focal sums)
    v2f ones; ones.x = 1.0f; ones.y = 1.0f;
    float box_acc = 0.0f;
    float pos_acc = 0.0f;

    const int   Ws[3]  = {80, 40, 20};
    const int   Hs[3]  = {80, 40, 20};
    const float inv[3] = {1.0f / 8.0f, 1.0f / 16.0f, 1.0f / 32.0f};
    int offset = 0;
    #pragma unroll
    for (int lvl = 0; lvl < 3; ++lvl) {
        const int W = Ws[lvl], H = Hs[lvl];
        const int gi = (int)floorf(cx * inv[lvl]);
        const int gj = (int)floorf(cy * inv[lvl]);
        const bool valid = gmask && (gi >= 0) && (gj >= 0) && (gi < W) && (gj < H);
        const int  idx   = valid ? (offset + gj * W + gi) : 0;
        const float vf   = valid ? 1.0f : 0.0f;
        const size_t row = (size_t)b * NN + (size_t)idx;

        // ---- GIoU (once per pair: gated, not branched, to keep EXEC full) --
        const float4 pb = *(const float4*)(box_preds + row * 4);
        float ix1 = fmaxf(pb.x, gb.x), iy1 = fmaxf(pb.y, gb.y);
        float ix2 = fminf(pb.z, gb.z), iy2 = fminf(pb.w, gb.w);
        float inter = fmaxf(ix2 - ix1, 0.0f) * fmaxf(iy2 - iy1, 0.0f);
        float parea = (pb.z - pb.x) * (pb.w - pb.y);
        float uni   = parea + garea - inter;
        float iou   = inter * __builtin_amdgcn_rcpf(uni);
        float ex1 = fminf(pb.x, gb.x), ey1 = fminf(pb.y, gb.y);
        float ex2 = fmaxf(pb.z, gb.z), ey2 = fmaxf(pb.w, gb.w);
        float encl = (ex2 - ex1) * (ey2 - ey1);
        float giou = iou - (encl - uni) * __builtin_amdgcn_rcpf(encl);
        box_acc += gate * vf * (1.0f - giou);
        pos_acc += gate * vf;

        // ---- focal loss over 80 classes via WMMA f32 16x16x4, B = ones ----
        // A layout (ISA 7.12.2): lanes 0-15 supply K={0,1}, lanes 16-31
        // supply K={2,3} of the same 16 rows -> lane half picks class pair.
        const float* crow = cls_preds + row * NUM_CLASSES;
        #pragma unroll
        for (int kb = 0; kb < NUM_CLASSES; kb += 4) {
            const int c0 = kb + (half << 1);
            const float2 x2 = *(const float2*)(crow + c0);
            v2f a;
            a.x = vf * focal_term(x2.x, c0 == label);
            a.y = vf * focal_term(x2.y, (c0 + 1) == label);
            cacc = __builtin_amdgcn_wmma_f32_16x16x4_f32(
                false, a, false, ones, (short)0, cacc, false, false);
        }
        offset += H * W;
    }

    // D columns are replicated (B = ones): lanes 0-15 hold rows 0-7 in
    // cacc[0..7], lanes 16-31 hold rows 8-15.  Sum rows per lane, then fold
    // the two halves -> every lane holds this wave's 16-pair cls total.
    float cls_half = cacc[0] + cacc[1] + cacc[2] + cacc[3] +
                     cacc[4] + cacc[5] + cacc[6] + cacc[7];
    float cls_acc = cls_half + __shfl_xor(cls_half, 16, 32);

    #pragma unroll
    for (int d = 16; d >= 1; d >>= 1) {
        box_acc += __shfl_xor(box_acc, d, 32);
        pos_acc += __shfl_xor(pos_acc, d, 32);
    }

    __shared__ float s_box[WAVES_PER_BLOCK];
    __shared__ float s_cls[WAVES_PER_BLOCK];
    __shared__ float s_pos[WAVES_PER_BLOCK];
    if (lane == 0) { s_box[wid] = box_acc; s_cls[wid] = cls_acc; s_pos[wid] = pos_acc; }
    __syncthreads();
    if (threadIdx.x == 0) {
        float tb = 0.0f, tc = 0.0f, tp = 0.0f;
        #pragma unroll
        for (int i = 0; i < WAVES_PER_BLOCK; ++i) { tb += s_box[i]; tc += s_cls[i]; tp += s_pos[i]; }
        partials[blockIdx.x * 3 + 0] = tb;
        partials[blockIdx.x * 3 + 1] = tc;
        partials[blockIdx.x * 3 + 2] = tp;
    }
}

__global__ __launch_bounds__(32)
void yolo_loss_stage2(const float* __restrict__ partials, int nblocks,
                      float* __restrict__ out) {
    const int l = threadIdx.x;
    float tb = 0.0f, tc = 0.0f, tp = 0.0f;
    for (int i = l; i < nblocks; i += 32) {
        tb += partials[i * 3 + 0];
        tc += partials[i * 3 + 1];
        tp += partials[i * 3 + 2];
    }
    #pragma unroll
    for (int d = 16; d >= 1; d >>= 1) {
        tb += __shfl_xor(tb, d, 32);
        tc += __shfl_xor(tc, d, 32);
        tp += __shfl_xor(tp, d, 32);
    }
    if (l == 0) out[0] = (5.0f * tb + 1.0f * tc) / fmaxf(tp, 1.0f);
}

extern "C" void kernel_launch(void* const* d_in, const int* in_sizes, int n_in,
                              void* d_out, int out_size, void* d_ws, size_t ws_size,
                              hipStream_t stream) {
    (void)in_sizes; (void)n_in; (void)out_size; (void)ws_size;
    const float* box_preds = (const float*)d_in[0];
    const float* cls_preds = (const float*)d_in[1];
    const float* gt_boxes  = (const float*)d_in[2];
    const int*   gt_labels = (const int*)d_in[3];
    const unsigned char* gt_mask = (const unsigned char*)d_in[4];  // jnp bool8
    float* partials = (float*)d_ws;     // NUM_BLOCKS * 3 floats
    float* out      = (float*)d_out;

    yolo_loss_stage1<<<NUM_BLOCKS, BLOCK_THREADS, 0, stream>>>(
        box_preds, cls_preds, gt_boxes, gt_labels, gt_mask, partials);
    yolo_loss_stage2<<<1, 32, 0, stream>>>(partials, NUM_BLOCKS, out);
}